// OptNet_73813307949598
// MI455X (gfx1250) — compile-verified
//
#include <hip/hip_runtime.h>
#include <hip/hip_bf16.h>

// ---------------------------------------------------------------------------
// OptNet forward on MI455X (gfx1250, wave32).
// - All matrix-shaped work uses V_WMMA_F32_16X16X4_F32 (f32 precision required
//   by the interior-point solve; workload is L2-resident and compute bound).
// - The dominant kernel (per-sample Schur complement H = Q + G^T D G) streams
//   G into LDS with the Tensor Data Mover (tensor_load_to_lds), double-
//   buffered and synchronized with s_wait_tensorcnt, overlapping DMA with WMMA.
// ---------------------------------------------------------------------------

typedef float v2f __attribute__((ext_vector_type(2)));
typedef float v8f __attribute__((ext_vector_type(8)));
typedef unsigned int v4u __attribute__((ext_vector_type(4)));
typedef int v4i __attribute__((ext_vector_type(4)));
typedef int v8i __attribute__((ext_vector_type(8)));

__device__ __forceinline__ v8f wmma4(v2f a, v2f b, v8f c) {
    // D = A(16x4) * B(4x16) + C(16x16), f32
    return __builtin_amdgcn_wmma_f32_16x16x4_f32(
        /*neg_a=*/false, a, /*neg_b=*/false, b,
        /*c_mod=*/(short)0, c, /*reuse_a=*/false, /*reuse_b=*/false);
}

#define BSZ   2048
#define MCON  200
#define NVAR  64
#define IPM_ITERS 20
#define CH_ROWS 40          // G chunk rows per TDM transfer (5 chunks of 40)
#define CH_STRIDE 65        // LDS row stride (64 + 1 pad DWORD via TDM padding)

// Issue one TDM 2D tile load: G rows [c*40, c*40+40) x 64 cols -> LDS, with
// 1-DWORD padding after each 64-DWORD row (LDS row stride = 65 floats).
__device__ __forceinline__ void tdm_load_chunk(const float* G, int c,
                                               unsigned int lds_byte_addr) {
    const unsigned long long ga =
        (unsigned long long)(uintptr_t)(G + (size_t)c * CH_ROWS * 64);
    v4u g0;
    g0[0] = 1u;                                    // count=1, is_restore=0
    g0[1] = lds_byte_addr;                         // lds_addr (bytes)
    g0[2] = (unsigned int)(ga & 0xffffffffu);      // global_addr[31:0]
    g0[3] = (unsigned int)((ga >> 32) & 0x01ffffffu) | (2u << 30); // addr[56:32] | type=2
    v8i g1;
    g1[0] = (int)((2u << 16)    // data_size = 4 bytes
                | (1u << 20)    // pad_enable
                | (5u << 22)    // pad_interval: 64 DWORDs
                | (0u << 25));  // pad_amount: 1 DWORD
    g1[1] = (int)(64u << 16);   // tensor_dim0[15:0] = 64 (at bits 63:48)
    g1[2] = (int)((unsigned)CH_ROWS << 16); // tensor_dim0 hi = 0 | tensor_dim1 lo = 40
    g1[3] = (int)(64u << 16);   // tensor_dim1 hi = 0 | tile_dim0 = 64
    g1[4] = CH_ROWS;            // tile_dim1 = 40, tile_dim2 = 0 (2D tile)
    g1[5] = 64;                 // tensor_dim0_stride[31:0] = 64 elements
    g1[6] = 0;                  // stride hi | tensor_dim1_stride lo
    g1[7] = 0;
    v4i g2;
    g2[0] = 1; g2[1] = 0; g2[2] = 0; g2[3] = 0;    // tensor_dim2 = 1 (safe bound)
    v4i g3;
    g3[0] = 0; g3[1] = 0; g3[2] = 0; g3[3] = 0;
#if defined(__clang_major__) && (__clang_major__ >= 23)
    v8i gx;
    gx[0] = 0; gx[1] = 0; gx[2] = 0; gx[3] = 0;
    gx[4] = 0; gx[5] = 0; gx[6] = 0; gx[7] = 0;
    __builtin_amdgcn_tensor_load_to_lds(g0, g1, g2, g3, gx, 0);
#else
    __builtin_amdgcn_tensor_load_to_lds(g0, g1, g2, g3, 0);
#endif
}

// ---------------------------------------------------------------------------
// C[B,N] = relu(A[B,K] @ W[N,K]^T + bias[N]); one 16x16 tile per wave.
// ---------------------------------------------------------------------------
__global__ void __launch_bounds__(256)
k_gemm_relu(const float* __restrict__ A, const float* __restrict__ W,
            const float* __restrict__ bias, float* __restrict__ C,
            int K, int N) {
    const int lane = threadIdx.x & 31;
    const int wave = threadIdx.x >> 5;
    const int tile = blockIdx.x * 8 + wave;
    const int ntn  = N >> 4;
    const int I = tile / ntn;
    const int J = tile % ntn;
    const int m = lane & 15;
    const int half = lane >> 4;
    const float* Arow = A + (size_t)((I << 4) + m) * K;
    const float* Wrow = W + (size_t)((J << 4) + m) * K;  // B frag: n = lane&15
    v8f acc = {};
    for (int k = 0; k < K; k += 4) {
        const int kk = k + 2 * half;
        v2f a, b;
        a.x = Arow[kk];     a.y = Arow[kk + 1];
        b.x = Wrow[kk];     b.y = Wrow[kk + 1];
        acc = wmma4(a, b, acc);
    }
    const int col = (J << 4) + m;
    const float bv = bias[col];
#pragma unroll
    for (int g = 0; g < 8; ++g) {
        const int r = (I << 4) + g + half * 8;
        float v = acc[g] + bv;
        C[(size_t)r * N + col] = v > 0.f ? v : 0.f;
    }
}

// ---------------------------------------------------------------------------
// BatchNorm helpers: zero, partial column sums (atomic), apply.
// ---------------------------------------------------------------------------
__global__ void k_zero(float* __restrict__ p, int n) {
    int i = blockIdx.x * 256 + threadIdx.x;
    if (i < n) p[i] = 0.f;
}

__global__ void __launch_bounds__(256)
k_colsum(const float* __restrict__ X, float* __restrict__ sum,
         float* __restrict__ sumsq, int N) {
    const int col = blockIdx.x * 256 + threadIdx.x;
    if (col >= N) return;
    const int rows = BSZ >> 4;            // 16 row chunks (gridDim.y)
    const int r0 = blockIdx.y * rows;
    float s = 0.f, q = 0.f;
    for (int r = r0; r < r0 + rows; ++r) {
        float v = X[(size_t)r * N + col];
        s += v; q += v * v;
    }
    atomicAdd(&sum[col], s);
    atomicAdd(&sumsq[col], q);
}

__global__ void __launch_bounds__(256)
k_bn_apply(float* __restrict__ X, const float* __restrict__ sum,
           const float* __restrict__ sumsq, const float* __restrict__ gamma,
           const float* __restrict__ beta, int N) {
    const int idx = blockIdx.x * 256 + threadIdx.x;
    if (idx >= BSZ * N) return;
    const int col = idx % N;
    const float mean = sum[col] * (1.f / BSZ);
    const float var  = sumsq[col] * (1.f / BSZ) - mean * mean;
    const float inv  = rsqrtf(var + 1e-5f);
    X[idx] = gamma[col] * (X[idx] - mean) * inv + beta[col];
}

// ---------------------------------------------------------------------------
// Q = tril(L) tril(L)^T + 1e-4 I ; h = G z0 + s0  (one block)
// ---------------------------------------------------------------------------
__global__ void __launch_bounds__(256)
k_prep(const float* __restrict__ L, const float* __restrict__ G,
       const float* __restrict__ z0, const float* __restrict__ s0,
       float* __restrict__ Q, float* __restrict__ hvec) {
    const int t = threadIdx.x;
    for (int e = t; e < NVAR * NVAR; e += 256) {
        const int i = e >> 6, j = e & 63;
        const int mn = i < j ? i : j;
        float acc = 0.f;
        for (int k = 0; k <= mn; ++k) acc += L[i * 64 + k] * L[j * 64 + k];
        if (i == j) acc += 1e-4f;
        Q[e] = acc;
    }
    if (t < MCON) {
        float acc = s0[t];
        for (int j = 0; j < NVAR; ++j) acc += G[t * 64 + j] * z0[j];
        hvec[t] = acc;
    }
}

__global__ void k_init_state(float* __restrict__ z, float* __restrict__ s,
                             float* __restrict__ lam) {
    const int i = blockIdx.x * 256 + threadIdx.x;
    if (i < BSZ * NVAR) z[i] = 0.f;
    if (i < BSZ * MCON) { s[i] = 1.f; lam[i] = 1.f; }
}

// ---------------------------------------------------------------------------
// rz[B,64] = z@Q + lam@G + p      (WMMA, K = 64 then 200)
// ---------------------------------------------------------------------------
__global__ void __launch_bounds__(256)
k_rz(const float* __restrict__ z, const float* __restrict__ Q,
     const float* __restrict__ lam, const float* __restrict__ G,
     const float* __restrict__ p, float* __restrict__ rz) {
    const int lane = threadIdx.x & 31, wave = threadIdx.x >> 5;
    const int tile = blockIdx.x * 8 + wave;   // 128x4 = 512 tiles
    const int I = tile >> 2, J = tile & 3;
    const int m = lane & 15, half = lane >> 4;
    const float* zr = z + (size_t)((I << 4) + m) * NVAR;
    const float* lr = lam + (size_t)((I << 4) + m) * MCON;
    const int n = (J << 4) + m;
    v8f acc = {};
    for (int k = 0; k < NVAR; k += 4) {
        const int kk = k + 2 * half;
        v2f a, b;
        a.x = zr[kk];            a.y = zr[kk + 1];
        b.x = Q[kk * 64 + n];    b.y = Q[(kk + 1) * 64 + n];
        acc = wmma4(a, b, acc);
    }
    for (int k = 0; k < MCON; k += 4) {
        const int kk = k + 2 * half;
        v2f a, b;
        a.x = lr[kk];            a.y = lr[kk + 1];
        b.x = G[kk * 64 + n];    b.y = G[(kk + 1) * 64 + n];
        acc = wmma4(a, b, acc);
    }
#pragma unroll
    for (int g = 0; g < 8; ++g) {
        const int r = (I << 4) + g + half * 8;
        rz[(size_t)r * NVAR + n] = acc[g] + p[(size_t)r * NVAR + n];
    }
}

// ---------------------------------------------------------------------------
// rp[B,200] = z@G^T + s - h       (WMMA, N padded 200 -> 208)
// ---------------------------------------------------------------------------
__global__ void __launch_bounds__(256)
k_rp(const float* __restrict__ z, const float* __restrict__ G,
     const float* __restrict__ s, const float* __restrict__ hvec,
     float* __restrict__ rp) {
    const int lane = threadIdx.x & 31, wave = threadIdx.x >> 5;
    const int tile = blockIdx.x * 8 + wave;   // 128x13 = 1664 tiles = 208 blocks
    const int I = tile / 13, J = tile % 13;
    const int m = lane & 15, half = lane >> 4;
    const float* zr = z + (size_t)((I << 4) + m) * NVAR;
    const int n = (J << 4) + m;               // 0..207
    const bool nok = n < MCON;
    const float* Gn = G + (size_t)(nok ? n : 0) * 64;  // G^T(k,n) = G[n][k]
    v8f acc = {};
    for (int k = 0; k < NVAR; k += 4) {
        const int kk = k + 2 * half;
        v2f a, b;
        a.x = zr[kk];  a.y = zr[kk + 1];
        b.x = Gn[kk];  b.y = Gn[kk + 1];
        acc = wmma4(a, b, acc);
    }
    if (nok) {
        const float hv = hvec[n];
#pragma unroll
        for (int g = 0; g < 8; ++g) {
            const int r = (I << 4) + g + half * 8;
            rp[(size_t)r * MCON + n] = acc[g] + s[(size_t)r * MCON + n] - hv;
        }
    }
}

// ---------------------------------------------------------------------------
// per-sample: mu = 0.1*mean(s*lam); d = lam/s; w = (mu - s*lam + lam*rp)/s
// ---------------------------------------------------------------------------
__global__ void __launch_bounds__(256)
k_ipm_elem(const float* __restrict__ s, const float* __restrict__ lam,
           const float* __restrict__ rp, float* __restrict__ dvec,
           float* __restrict__ wvec, float* __restrict__ muvec) {
    __shared__ float red[256];
    const int b = blockIdx.x, t = threadIdx.x;
    float sv = 0.f, lv = 0.f, rv = 0.f, prod = 0.f;
    if (t < MCON) {
        sv = s[b * MCON + t]; lv = lam[b * MCON + t]; rv = rp[b * MCON + t];
        prod = sv * lv;
    }
    red[t] = prod;
    __syncthreads();
    for (int o = 128; o > 0; o >>= 1) {
        if (t < o) red[t] += red[t + o];
        __syncthreads();
    }
    const float mu = 0.1f * red[0] * (1.f / MCON);
    if (t == 0) muvec[b] = mu;
    if (t < MCON) {
        dvec[b * MCON + t] = lv / sv;
        wvec[b * MCON + t] = (mu - sv * lv + lv * rv) / sv;
    }
}

// ---------------------------------------------------------------------------
// rhs[B,64] = -(rz + w@G)          (WMMA, K = 200)
// ---------------------------------------------------------------------------
__global__ void __launch_bounds__(256)
k_rhs(const float* __restrict__ w, const float* __restrict__ G,
      const float* __restrict__ rz, float* __restrict__ rhs) {
    const int lane = threadIdx.x & 31, wave = threadIdx.x >> 5;
    const int tile = blockIdx.x * 8 + wave;   // 512 tiles
    const int I = tile >> 2, J = tile & 3;
    const int m = lane & 15, half = lane >> 4;
    const float* wr = w + (size_t)((I << 4) + m) * MCON;
    const int n = (J << 4) + m;
    v8f acc = {};
    for (int k = 0; k < MCON; k += 4) {
        const int kk = k + 2 * half;
        v2f a, b;
        a.x = wr[kk];            a.y = wr[kk + 1];
        b.x = G[kk * 64 + n];    b.y = G[(kk + 1) * 64 + n];
        acc = wmma4(a, b, acc);
    }
#pragma unroll
    for (int g = 0; g < 8; ++g) {
        const int r = (I << 4) + g + half * 8;
        rhs[(size_t)r * NVAR + n] = -(acc[g] + rz[(size_t)r * NVAR + n]);
    }
}

// ---------------------------------------------------------------------------
// Fused heavy kernel, one block per sample:
//   H = Q + G^T diag(d) G   (WMMA; G streamed into LDS by the TDM,
//                            double-buffered 40-row chunks)
//   Cholesky(H) in LDS, solve H dz = rhs
//   ds, dlam, alpha (fraction-to-boundary), update z/s/lam
// ---------------------------------------------------------------------------
__global__ void __launch_bounds__(256)
k_hsolve(const float* __restrict__ G, const float* __restrict__ Q,
         const float* __restrict__ dvec, const float* __restrict__ rhs,
         const float* __restrict__ rp, const float* __restrict__ muvec,
         float* __restrict__ z, float* __restrict__ s,
         float* __restrict__ lam) {
    __shared__ float Gs[2][CH_ROWS * CH_STRIDE];   // double-buffered G chunks
    __shared__ float Hl[64 * 65];
    __shared__ float dl[MCON];
    __shared__ float xv[64];
    __shared__ float red[256];
    const int b = blockIdx.x;
    const int t = threadIdx.x;
    const int lane = t & 31, wave = t >> 5;
    const int m = lane & 15, half = lane >> 4;

    if (t < MCON) dl[t] = dvec[b * MCON + t];

    // each wave owns tiles (wave) and (wave+8) of the 4x4 tile grid
    const int I0 = wave >> 2, J0 = wave & 3;
    const int I1 = (wave + 8) >> 2, J1 = (wave + 8) & 3;
    const int cA0 = I0 * 16 + m, cB0 = J0 * 16 + m;
    const int cA1 = I1 * 16 + m, cB1 = J1 * 16 + m;
    v8f acc0 = {}, acc1 = {};

    const unsigned int lds0 = (unsigned int)(uintptr_t)&Gs[0][0];
    const unsigned int lds1 = (unsigned int)(uintptr_t)&Gs[1][0];

    if (wave == 0) tdm_load_chunk(G, 0, lds0);
    for (int c = 0; c < 5; ++c) {
        if (wave == 0) {
            if (c + 1 < 5) {
                tdm_load_chunk(G, c + 1, ((c + 1) & 1) ? lds1 : lds0);
                __builtin_amdgcn_s_wait_tensorcnt(1);   // chunk c complete
            } else {
                __builtin_amdgcn_s_wait_tensorcnt(0);   // last chunk complete
            }
        }
        __syncthreads();                 // chunk c visible to all waves
        const float* Gc = &Gs[c & 1][0];
        const int k0 = c * CH_ROWS;
        for (int kk = 0; kk < CH_ROWS; kk += 4) {
            const int kl = kk + 2 * half;
            const float d0 = dl[k0 + kl];
            const float d1 = dl[k0 + kl + 1];
            v2f a, bf;
            a.x  = Gc[kl * CH_STRIDE + cA0] * d0;
            a.y  = Gc[(kl + 1) * CH_STRIDE + cA0] * d1;
            bf.x = Gc[kl * CH_STRIDE + cB0];
            bf.y = Gc[(kl + 1) * CH_STRIDE + cB0];
            acc0 = wmma4(a, bf, acc0);
            a.x  = Gc[kl * CH_STRIDE + cA1] * d0;
            a.y  = Gc[(kl + 1) * CH_STRIDE + cA1] * d1;
            bf.x = Gc[kl * CH_STRIDE + cB1];
            bf.y = Gc[(kl + 1) * CH_STRIDE + cB1];
            acc1 = wmma4(a, bf, acc1);
        }
        __syncthreads();                 // all done with buf (c&1) before reuse
    }

    // H = acc + Q into LDS
#pragma unroll
    for (int g = 0; g < 8; ++g) {
        const int r0 = I0 * 16 + g + half * 8, c0 = J0 * 16 + m;
        Hl[r0 * 65 + c0] = acc0[g] + Q[r0 * 64 + c0];
        const int r1 = I1 * 16 + g + half * 8, c1 = J1 * 16 + m;
        Hl[r1 * 65 + c1] = acc1[g] + Q[r1 * 64 + c1];
    }
    if (t < 64) xv[t] = rhs[b * NVAR + t];
    __syncthreads();

    // in-LDS Cholesky (lower)
    for (int j = 0; j < 64; ++j) {
        if (t == 0) Hl[j * 65 + j] = sqrtf(Hl[j * 65 + j]);
        __syncthreads();
        if (t > j && t < 64) Hl[t * 65 + j] /= Hl[j * 65 + j];
        __syncthreads();
        if (t > j && t < 64) {
            const float lij = Hl[t * 65 + j];
            for (int k = j + 1; k <= t; ++k)
                Hl[t * 65 + k] -= lij * Hl[k * 65 + j];
        }
        __syncthreads();
    }
    // forward: L y = rhs
    for (int j = 0; j < 64; ++j) {
        if (t == 0) xv[j] /= Hl[j * 65 + j];
        __syncthreads();
        if (t > j && t < 64) xv[t] -= Hl[t * 65 + j] * xv[j];
        __syncthreads();
    }
    // backward: L^T dz = y
    for (int j = 63; j >= 0; --j) {
        if (t == 0) xv[j] /= Hl[j * 65 + j];
        __syncthreads();
        if (t < j) xv[t] -= Hl[j * 65 + t] * xv[j];
        __syncthreads();
    }

    // ds, dlam, fraction-to-boundary alpha, update
    const float mu = muvec[b];
    float sv = 1.f, lv = 1.f, dsk = 0.f, dlk = 0.f, lim = 1e30f;
    if (t < MCON) {
        sv = s[b * MCON + t];
        lv = lam[b * MCON + t];
        const float* Gr = G + t * 64;
        float dot = 0.f;
#pragma unroll 8
        for (int j = 0; j < 64; ++j) dot += Gr[j] * xv[j];
        dsk = -rp[b * MCON + t] - dot;
        dlk = (mu - sv * lv - lv * dsk) / sv;
        const float a1 = dsk < 0.f ? -sv / dsk : 1e30f;
        const float a2 = dlk < 0.f ? -lv / dlk : 1e30f;
        lim = fminf(a1, a2);
    }
    red[t] = lim;
    __syncthreads();
    for (int o = 128; o > 0; o >>= 1) {
        if (t < o) red[t] = fminf(red[t], red[t + o]);
        __syncthreads();
    }
    const float alpha = 0.99f * fminf(1.0f, red[0]);
    if (t < MCON) {
        s[b * MCON + t]   = sv + alpha * dsk;
        lam[b * MCON + t] = lv + alpha * dlk;
    }
    if (t < 64) z[b * NVAR + t] += alpha * xv[t];
}

// ---------------------------------------------------------------------------
// log_softmax over rows of z[B,64]
// ---------------------------------------------------------------------------
__global__ void __launch_bounds__(64)
k_logsoftmax(const float* __restrict__ z, float* __restrict__ out) {
    __shared__ float red[64];
    const int b = blockIdx.x, t = threadIdx.x;
    const float v = z[b * NVAR + t];
    red[t] = v;
    __syncthreads();
    for (int o = 32; o > 0; o >>= 1) {
        if (t < o) red[t] = fmaxf(red[t], red[t + o]);
        __syncthreads();
    }
    const float mx = red[0];
    __syncthreads();
    red[t] = expf(v - mx);
    __syncthreads();
    for (int o = 32; o > 0; o >>= 1) {
        if (t < o) red[t] += red[t + o];
        __syncthreads();
    }
    out[b * NVAR + t] = (v - mx) - logf(red[0]);
}

// ---------------------------------------------------------------------------
extern "C" void kernel_launch(void* const* d_in, const int* in_sizes, int n_in,
                              void* d_out, int out_size, void* d_ws, size_t ws_size,
                              hipStream_t stream) {
    const float* x    = (const float*)d_in[0];
    const float* W1   = (const float*)d_in[1];
    const float* b1   = (const float*)d_in[2];
    const float* W2   = (const float*)d_in[3];
    const float* b2   = (const float*)d_in[4];
    const float* bn1g = (const float*)d_in[5];
    const float* bn1b = (const float*)d_in[6];
    const float* bn2g = (const float*)d_in[7];
    const float* bn2b = (const float*)d_in[8];
    const float* L    = (const float*)d_in[9];
    const float* G    = (const float*)d_in[10];
    const float* z0   = (const float*)d_in[11];
    const float* s0   = (const float*)d_in[12];
    float* out = (float*)d_out;

    float* ws = (float*)d_ws;
    size_t off = 0;
    float* h1  = ws + off; off += (size_t)BSZ * 1024;
    float* p   = ws + off; off += (size_t)BSZ * NVAR;
    float* Qm  = ws + off; off += 64 * 64;
    float* hv  = ws + off; off += 256;
    float* zv  = ws + off; off += (size_t)BSZ * NVAR;
    float* sv  = ws + off; off += (size_t)BSZ * MCON;
    float* lv  = ws + off; off += (size_t)BSZ * MCON;
    float* rzv = ws + off; off += (size_t)BSZ * NVAR;
    float* rpv = ws + off; off += (size_t)BSZ * MCON;
    float* dv  = ws + off; off += (size_t)BSZ * MCON;
    float* wv  = ws + off; off += (size_t)BSZ * MCON;
    float* muv = ws + off; off += BSZ;
    float* rhv = ws + off; off += (size_t)BSZ * NVAR;
    float* st  = ws + off; off += 4096;   // sum1|sq1|sum2|sq2
    float* sum1 = st, *sq1 = st + 1024, *sum2 = st + 2048, *sq2 = st + 2048 + 64;

    // layer 1: h1 = relu(x @ W1^T + b1), then batchnorm
    k_gemm_relu<<<(128 * 64) / 8, 256, 0, stream>>>(x, W1, b1, h1, 1024, 1024);
    k_zero<<<16, 256, 0, stream>>>(st, 4096);
    {
        dim3 g1((1024 + 255) / 256, 16);
        k_colsum<<<g1, 256, 0, stream>>>(h1, sum1, sq1, 1024);
    }
    k_bn_apply<<<(BSZ * 1024 + 255) / 256, 256, 0, stream>>>(h1, sum1, sq1, bn1g, bn1b, 1024);

    // layer 2: p = relu(h1 @ W2^T + b2), then batchnorm
    k_gemm_relu<<<(128 * 4) / 8, 256, 0, stream>>>(h1, W2, b2, p, 1024, 64);
    {
        dim3 g2(1, 16);
        k_colsum<<<g2, 256, 0, stream>>>(p, sum2, sq2, 64);
    }
    k_bn_apply<<<(BSZ * 64 + 255) / 256, 256, 0, stream>>>(p, sum2, sq2, bn2g, bn2b, 64);

    // QP setup
    k_prep<<<1, 256, 0, stream>>>(L, G, z0, s0, Qm, hv);
    k_init_state<<<(BSZ * MCON + 255) / 256, 256, 0, stream>>>(zv, sv, lv);

    // interior-point iterations
    for (int it = 0; it < IPM_ITERS; ++it) {
        k_rz<<<64, 256, 0, stream>>>(zv, Qm, lv, G, p, rzv);
        k_rp<<<208, 256, 0, stream>>>(zv, G, sv, hv, rpv);
        k_ipm_elem<<<BSZ, 256, 0, stream>>>(sv, lv, rpv, dv, wv, muv);
        k_rhs<<<64, 256, 0, stream>>>(wv, G, rzv, rhv);
        k_hsolve<<<BSZ, 256, 0, stream>>>(G, Qm, dv, rhv, rpv, muv, zv, sv, lv);
    }

    k_logsoftmax<<<BSZ, 64, 0, stream>>>(zv, out);
}